// DynTCN_11982958756055
// MI455X (gfx1250) — compile-verified
//
#include <hip/hip_runtime.h>

// CDNA5 / gfx1250. wave32. V_WMMA_F32_16X16X4_F32 for all dense GEMMs (f32
// matches the f32 reference), HW f32 atomics for the graph scatter (10MB
// accumulator resident in 192MB L2), algebraic folding of TCN+proj into three
// 128x128 matrices, and a loop-invariant decoder GCN (computed once).

typedef __attribute__((ext_vector_type(2))) float v2f;
typedef __attribute__((ext_vector_type(8))) float v8f;

#define HDIM 128
#define FDIM 16

// ---------------------------------------------------------------- utilities
__global__ void zero_f32(float* __restrict__ p, int n) {
    int i = blockIdx.x * blockDim.x + threadIdx.x;
    if (i < n) p[i] = 0.0f;
}

// Fold conv taps that survive at the last temporal position through proj_w:
//   h_new = T1@M1 + T0@M0 + hg@Mg + biasTot
__global__ void precompute_mats(const float* __restrict__ cw, const float* __restrict__ pw,
                                const float* __restrict__ cb, const float* __restrict__ pb,
                                float* __restrict__ M1, float* __restrict__ M0,
                                float* __restrict__ Mg, float* __restrict__ biasTot) {
    int i = blockIdx.x;    // input channel (K-dim row of M*)
    int h = threadIdx.x;   // output channel
    float m1 = 0.f, m0 = 0.f, mg = 0.f;
    for (int o = 0; o < HDIM; ++o) {
        float p0 = pw[h * 384 + o];
        float p1 = pw[h * 384 + 128 + o];
        float p2 = pw[h * 384 + 256 + o];
        m1 += cw[((0 * HDIM + o) * HDIM + i) * 3 + 0] * p0;
        m0 += cw[((1 * HDIM + o) * HDIM + i) * 3 + 0] * p1;
        mg += cw[((0 * HDIM + o) * HDIM + i) * 3 + 1] * p0
            + cw[((1 * HDIM + o) * HDIM + i) * 3 + 1] * p1
            + cw[((2 * HDIM + o) * HDIM + i) * 3 + 1] * p2;
    }
    M1[i * HDIM + h] = m1;
    M0[i * HDIM + h] = m0;
    Mg[i * HDIM + h] = mg;
    if (i == 0) {
        float b = pb[h];
        for (int o = 0; o < HDIM; ++o) {
            b += cb[0 * HDIM + o] * pw[h * 384 + o]
               + cb[1 * HDIM + o] * pw[h * 384 + 128 + o]
               + cb[2 * HDIM + o] * pw[h * 384 + 256 + o];
        }
        biasTot[h] = b;
    }
}

// ---------------------------------------------------------------- GCN stage
__global__ void deg_init(float* __restrict__ deg, int n) {
    int i = blockIdx.x * blockDim.x + threadIdx.x;
    if (i < n) deg[i] = 1.0f;   // self-loop
}

__global__ void deg_scatter(const long long* __restrict__ dst, float* __restrict__ deg, int e) {
    int i = blockIdx.x * blockDim.x + threadIdx.x;
    if (i < e) atomicAdd(&deg[(int)dst[i]], 1.0f);
}

__global__ void deg_rsqrt(float* __restrict__ deg, int n) {
    int i = blockIdx.x * blockDim.x + threadIdx.x;
    if (i < n) deg[i] = rsqrtf(deg[i]);   // in place -> dinv
}

// xw = x_t @ gcn_w   ([N,16] @ [16,128]) via WMMA f32 16x16x4, K=16 -> 4 steps.
__global__ void gemm_xw(const float* __restrict__ x, const float* __restrict__ gw,
                        float* __restrict__ xw, int N) {
    __shared__ float As[16 * FDIM];
    int n0  = blockIdx.x * 16;
    int tid = threadIdx.x;                 // 256 threads = 8 waves
    const bool full = (n0 + 16 <= N);      // uniform
    if (full) {
        if (tid < 64) ((float4*)As)[tid] = ((const float4*)(x + (size_t)n0 * FDIM))[tid];
    } else {
        int row = n0 + tid / FDIM;
        As[tid] = (row < N) ? x[(size_t)n0 * FDIM + tid] : 0.0f;
    }
    __syncthreads();
    int wave = tid >> 5, lane = tid & 31;
    int c0 = wave * 16;
    int hi = lane >> 4;      // half-wave select
    int r  = lane & 15;
    v8f acc = {0.f, 0.f, 0.f, 0.f, 0.f, 0.f, 0.f, 0.f};
    for (int kb = 0; kb < FDIM; kb += 4) {
        v2f a, b;
        a.x = As[r * FDIM + kb + (hi ? 2 : 0)];
        a.y = As[r * FDIM + kb + (hi ? 3 : 1)];
        b.x = gw[(kb + (hi ? 2 : 0)) * HDIM + c0 + r];
        b.y = gw[(kb + (hi ? 3 : 1)) * HDIM + c0 + r];
        acc = __builtin_amdgcn_wmma_f32_16x16x4_f32(false, a, false, b, (short)0, acc,
                                                    false, false);
    }
    float* outp = xw + (size_t)(n0 + (hi ? 8 : 0)) * HDIM + c0 + r;
    if (full) {
#pragma unroll
        for (int v = 0; v < 8; ++v) outp[(size_t)v * HDIM] = acc[v];
    } else {
        for (int v = 0; v < 8; ++v) {
            int row = n0 + v + (hi ? 8 : 0);
            if (row < N) xw[(size_t)row * HDIM + c0 + r] = acc[v];
        }
    }
}

// acc[n,h] = xw[n,h]*dinv[n]^2 + gcn_b[h]  (self-loop term + bias)
__global__ void acc_init(const float* __restrict__ xw, const float* __restrict__ dinv,
                         const float* __restrict__ gb, float* __restrict__ acc, int total) {
    int i = blockIdx.x * blockDim.x + threadIdx.x;
    if (i < total) {
        int n = i >> 7, h = i & 127;
        float d = dinv[n];
        acc[i] = xw[i] * d * d + gb[h];
    }
}

// one wave per edge; lane handles 4 floats of the H=128 row; f32 HW atomics to L2
__global__ void edge_scatter(const long long* __restrict__ src, const long long* __restrict__ dst,
                             const float* __restrict__ xw, const float* __restrict__ dinv,
                             float* __restrict__ acc, int E) {
    int e = blockIdx.x * 8 + (threadIdx.x >> 5);
    if (e >= E) return;
    int lane = threadIdx.x & 31;
    int s = (int)src[e], d = (int)dst[e];
    float c = dinv[s] * dinv[d];
    const float4 v = ((const float4*)(xw + (size_t)s * HDIM))[lane];
    float* ad = acc + (size_t)d * HDIM + lane * 4;
    atomicAdd(ad + 0, v.x * c);
    atomicAdd(ad + 1, v.y * c);
    atomicAdd(ad + 2, v.z * c);
    atomicAdd(ad + 3, v.w * c);
}

__global__ void relu_inplace(float* __restrict__ a, int n) {
    int i = blockIdx.x * blockDim.x + threadIdx.x;
    if (i < n) a[i] = fmaxf(a[i], 0.0f);
}

// ---------------------------------------------------- fused TCN+proj GEMM
// OUT = T1@M1 + T0@M0 + HG@Mg + biasTot    (all [N,128] @ [128,128])
__global__ void gemm_fused(const float* __restrict__ T1, const float* __restrict__ T0,
                           const float* __restrict__ HG,
                           const float* __restrict__ M1, const float* __restrict__ M0,
                           const float* __restrict__ Mg, const float* __restrict__ biasTot,
                           float* __restrict__ OUT, int N) {
    __shared__ float As[3][16 * HDIM];   // 24 KB
    int n0  = blockIdx.x * 16;
    int tid = threadIdx.x;               // 256 threads = 8 waves -> 16x128 tile
    const bool full = (n0 + 16 <= N);    // uniform

    // prefetch next block's A strips (global_prefetch_b8); 256 thr * 32B = 8KB
    {
        int pn = (blockIdx.x + 1) * 16;
        if (pn + 16 <= N) {
            size_t off = (size_t)pn * HDIM;
            __builtin_prefetch((const char*)(T1 + off) + tid * 32, 0, 3);
            __builtin_prefetch((const char*)(T0 + off) + tid * 32, 0, 3);
            __builtin_prefetch((const char*)(HG + off) + tid * 32, 0, 3);
        }
    }

    if (full) {
        const float4* t1 = (const float4*)(T1 + (size_t)n0 * HDIM);
        const float4* t0 = (const float4*)(T0 + (size_t)n0 * HDIM);
        const float4* hgp = (const float4*)(HG + (size_t)n0 * HDIM);
#pragma unroll
        for (int j = tid; j < 16 * HDIM / 4; j += 256) {   // 2 iterations
            ((float4*)As[0])[j] = t1[j];
            ((float4*)As[1])[j] = t0[j];
            ((float4*)As[2])[j] = hgp[j];
        }
    } else {
        for (int j = tid; j < 16 * HDIM; j += 256) {
            int row = n0 + j / HDIM;
            bool ok = row < N;
            As[0][j] = ok ? T1[(size_t)n0 * HDIM + j] : 0.0f;
            As[1][j] = ok ? T0[(size_t)n0 * HDIM + j] : 0.0f;
            As[2][j] = ok ? HG[(size_t)n0 * HDIM + j] : 0.0f;
        }
    }
    __syncthreads();
    int wave = tid >> 5, lane = tid & 31;
    int c0 = wave * 16;
    int hi = lane >> 4;
    int r  = lane & 15;
    v8f acc = {0.f, 0.f, 0.f, 0.f, 0.f, 0.f, 0.f, 0.f};
    const float* Bm[3] = {M1, M0, Mg};
    for (int m = 0; m < 3; ++m) {
        const float* B = Bm[m];
        const float* A = &As[m][0];
        for (int kb = 0; kb < HDIM; kb += 4) {
            v2f a, b;
            a.x = A[r * HDIM + kb + (hi ? 2 : 0)];
            a.y = A[r * HDIM + kb + (hi ? 3 : 1)];
            b.x = B[(kb + (hi ? 2 : 0)) * HDIM + c0 + r];
            b.y = B[(kb + (hi ? 3 : 1)) * HDIM + c0 + r];
            acc = __builtin_amdgcn_wmma_f32_16x16x4_f32(false, a, false, b, (short)0, acc,
                                                        false, false);
        }
    }
    float bb = biasTot[c0 + r];
    float* outp = OUT + (size_t)(n0 + (hi ? 8 : 0)) * HDIM + c0 + r;
    if (full) {
#pragma unroll
        for (int v = 0; v < 8; ++v) outp[(size_t)v * HDIM] = acc[v] + bb;
    } else {
        for (int v = 0; v < 8; ++v) {
            int row = n0 + v + (hi ? 8 : 0);
            if (row < N) OUT[(size_t)row * HDIM + c0 + r] = acc[v] + bb;
        }
    }
}

// preds = h_new @ head_w^T + head_b   ([N,128] @ [128,16])
__global__ void head_gemm(const float* __restrict__ hn, const float* __restrict__ hw,
                          const float* __restrict__ hb, float* __restrict__ out, int N) {
    __shared__ float Hs[16 * HDIM];
    int n0  = blockIdx.x * 16;
    int tid = threadIdx.x;   // 256
    const bool full = (n0 + 16 <= N);
    if (full) {
        const float4* hp = (const float4*)(hn + (size_t)n0 * HDIM);
#pragma unroll
        for (int j = tid; j < 16 * HDIM / 4; j += 256) ((float4*)Hs)[j] = hp[j];
    } else {
        for (int j = tid; j < 16 * HDIM; j += 256) {
            int row = n0 + j / HDIM;
            Hs[j] = (row < N) ? hn[(size_t)n0 * HDIM + j] : 0.0f;
        }
    }
    __syncthreads();
    int rl = tid >> 4;         // 0..15 local row
    int c  = tid & 15;         // 0..15 output channel
    int n  = n0 + rl;
    float s = hb[c];
    const float* hr = &Hs[rl * HDIM];
    for (int h = 0; h < HDIM; ++h) s += hr[h] * hw[c * HDIM + h];
    if (n < N) out[(size_t)n * 16 + c] = s;
}

// ---------------------------------------------------------------- launch
extern "C" void kernel_launch(void* const* d_in, const int* in_sizes, int n_in,
                              void* d_out, int out_size, void* d_ws, size_t ws_size,
                              hipStream_t stream) {
    const float*     x_seq  = (const float*)d_in[0];
    const long long* ei     = (const long long*)d_in[1];   // int64 [T,2,E]
    const float*     gcn_w  = (const float*)d_in[5];
    const float*     gcn_b  = (const float*)d_in[6];
    const float*     conv_w = (const float*)d_in[7];
    const float*     conv_b = (const float*)d_in[8];
    const float*     proj_w = (const float*)d_in[9];
    const float*     proj_b = (const float*)d_in[10];
    const float*     head_w = (const float*)d_in[11];
    const float*     head_b = (const float*)d_in[12];
    float*           outp   = (float*)d_out;

    const int HOR = 6, C = 16;
    const int N = out_size / (HOR * C);            // 20000
    const int T = in_sizes[4] / N;                 // 12   (id_seq = T*N)
    const int E = in_sizes[1] / (2 * T);           // 320000
    const int F = in_sizes[0] / in_sizes[4];       // 16
    const int NH = N * HDIM;

    // workspace partition (all f32): ~51.5 MB
    float* p   = (float*)d_ws;
    float* xw  = p; p += NH;          // x @ gcn_w
    float* hg  = p; p += NH;          // GCN accumulator -> relu'd hidden
    float* dinv= p; p += N;           // degree -> rsqrt(degree)
    float* w0  = p; p += NH;          // window ring (3 contiguous buffers)
    float* w1  = p; p += NH;
    float* w2  = p; p += NH;
    float* M1  = p; p += HDIM * HDIM;
    float* M0  = p; p += HDIM * HDIM;
    float* Mg  = p; p += HDIM * HDIM;
    float* bT  = p; p += HDIM;

    // zero window ring (contiguous)
    zero_f32<<<(3 * NH + 255) / 256, 256, 0, stream>>>(w0, 3 * NH);
    precompute_mats<<<HDIM, HDIM, 0, stream>>>(conv_w, proj_w, conv_b, proj_b, M1, M0, Mg, bT);

    float* win[3] = {w0, w1, w2};   // win[0]=oldest slot, win[2]=newest
    const int nBlk = (N + 15) / 16;

    // ---------------- encoder: 12 full cells ----------------
    for (int t = 0; t < T; ++t) {
        const long long* src = ei + (size_t)t * 2 * E;
        const long long* dst = src + E;
        deg_init   <<<(N + 255) / 256, 256, 0, stream>>>(dinv, N);
        deg_scatter<<<(E + 255) / 256, 256, 0, stream>>>(dst, dinv, E);
        deg_rsqrt  <<<(N + 255) / 256, 256, 0, stream>>>(dinv, N);
        gemm_xw    <<<nBlk, 256, 0, stream>>>(x_seq + (size_t)t * N * F, gcn_w, xw, N);
        acc_init   <<<(NH + 255) / 256, 256, 0, stream>>>(xw, dinv, gcn_b, hg, NH);
        edge_scatter<<<(E + 7) / 8, 256, 0, stream>>>(src, dst, xw, dinv, hg, E);
        relu_inplace<<<(NH + 255) / 256, 256, 0, stream>>>(hg, NH);
        // h_new written into the evicted slot win[0]; tcn slots: T0=win[1], T1=win[2]
        gemm_fused <<<nBlk, 256, 0, stream>>>(win[2], win[1], hg, M1, M0, Mg, bT, win[0], N);
        float* tmp = win[0]; win[0] = win[1]; win[1] = win[2]; win[2] = tmp;
    }

    // ---------------- decoder: GCN output hg is loop-invariant ----------------
    // (decoder input == x_seq[-1], edges == last edge set -> hg from t=T-1 reused)
    for (int s = 0; s < HOR; ++s) {
        gemm_fused<<<nBlk, 256, 0, stream>>>(win[2], win[1], hg, M1, M0, Mg, bT, win[0], N);
        head_gemm <<<nBlk, 256, 0, stream>>>(win[0], head_w, head_b, outp + (size_t)s * N * C, N);
        float* tmp = win[0]; win[0] = win[1]; win[1] = win[2]; win[2] = tmp;
    }
}